// SAGENet_33852932227159
// MI455X (gfx1250) — compile-verified
//
#include <hip/hip_runtime.h>

typedef __attribute__((ext_vector_type(2))) float v2f;
typedef __attribute__((ext_vector_type(8))) float v8f;

// Problem constants (deterministic from reference setup_inputs()).
constexpr int T      = 12;
constexpr int C_IN   = 32;
constexpr int H      = 64;
constexpr int C_OUT  = 32;
constexpr int N_SRC  = 50000;
constexpr int N_DST1 = 20000;
constexpr int N_DST2 = 10000;
constexpr int ROW    = T * C_IN;       // 384 floats per node "row" in (n,t,c) layout
constexpr int ROW4   = ROW / 4;        // 96 float4 per row
constexpr int M1     = N_DST1 * T;     // 240000 rows, K=32 -> H=64
constexpr int M2     = N_DST2 * T;     // 120000 rows, K=64 -> 32
constexpr float NEG_SLOPE = 0.01f;

__device__ __forceinline__ float leaky(float v) {
    return v > 0.f ? v : NEG_SLOPE * v;
}

// Load one f32 16x16x4 WMMA A/B fragment (float2 per lane).
// A (16x4 slice of row-major [M x ld]):  m = r0 + (lane&15), k = k0 + 2*(lane>>4)
// B (4x16 slice, B[k][n] = W[n][k], W row-major [N x ld]): identical indexing with n0.
__device__ __forceinline__ v2f load_frag(const float* __restrict__ base, int r0, int ld,
                                         int k0, int lane) {
    int r = r0 + (lane & 15);
    int k = k0 + ((lane >> 4) << 1);
    const float* p = base + (size_t)r * ld + k;
    v2f f;
    f.x = p[0];
    f.y = p[1];
    return f;
}

// ---------------------------------------------------------------- transpose
// node_feat (T, N_SRC, C_IN) -> xT (N_SRC, T, C_IN), float4 granularity.
__global__ void k_transpose(const float* __restrict__ nf, float* __restrict__ xT) {
    int idx = blockIdx.x * blockDim.x + threadIdx.x;      // float4 index into xT
    constexpr int C4 = C_IN / 4;
    constexpr int total = N_SRC * T * C4;
    if (idx >= total) return;
    int c4 = idx % C4;
    int t  = (idx / C4) % T;
    int n  = idx / (C4 * T);
    const float4* src = reinterpret_cast<const float4*>(nf);
    float4 v = src[(size_t)t * N_SRC * C4 + (size_t)n * C4 + c4];
    reinterpret_cast<float4*>(xT)[idx] = v;
}

// ---------------------------------------------------------------- CSR build
__global__ void k_zero_int(int* __restrict__ p, int n) {
    int i = blockIdx.x * blockDim.x + threadIdx.x;
    if (i < n) p[i] = 0;
}

__global__ void k_hist(const int* __restrict__ dst, int* __restrict__ counts, int E) {
    int e = blockIdx.x * blockDim.x + threadIdx.x;
    if (e < E) atomicAdd(&counts[dst[e]], 1);
}

// Single-block exclusive scan (N <= ~1M). Writes starts[0..N] and cursor[0..N-1].
__global__ void __launch_bounds__(1024)
k_scan(const int* __restrict__ counts, int* __restrict__ starts,
       int* __restrict__ cursor, int N) {
    __shared__ int part[1024];
    int tid   = threadIdx.x;
    int chunk = (N + 1023) >> 10;
    int begin = tid * chunk;
    int end   = begin + chunk;
    if (end > N) end = N;
    int sum = 0;
    for (int i = begin; i < end; ++i) sum += counts[i];
    part[tid] = sum;
    __syncthreads();
    for (int off = 1; off < 1024; off <<= 1) {
        int v = 0;
        if (tid >= off) v = part[tid - off];
        __syncthreads();
        part[tid] += v;
        __syncthreads();
    }
    int run = part[tid] - sum;                 // exclusive prefix for this chunk
    for (int i = begin; i < end; ++i) {
        starts[i] = run;
        cursor[i] = run;
        run += counts[i];
    }
    if (tid == 1023) starts[N] = part[1023];
}

// Scatter edges into dst-ordered buckets: (src, weight) pairs.
__global__ void k_bin(const int* __restrict__ src, const int* __restrict__ dst,
                      const float* __restrict__ ew, int* __restrict__ cursor,
                      int* __restrict__ srcs, float* __restrict__ ws, int E) {
    int e = blockIdx.x * blockDim.x + threadIdx.x;
    if (e >= E) return;
    int pos = atomicAdd(&cursor[dst[e]], 1);
    srcs[pos] = src[e];
    ws[pos]   = ew[e];
}

// ---------------------------------------------------------------- aggregation (gather)
// One block per dst node; 96 threads each own one float4 of the 384-float row.
// agg[n] = sum_j ws[j] * x[srcs[j]]  over j in [starts[n], starts[n+1]).
// Edge index/weight loads are block-uniform (scalarized); x reads are coalesced
// 1536B bursts that hit the MI455X 192MB L2. No atomics, no zero-fill pass.
__global__ void __launch_bounds__(96)
k_aggregate(const int* __restrict__ starts, const int* __restrict__ srcs,
            const float* __restrict__ ws, const float* __restrict__ x,
            float* __restrict__ agg) {
    int n   = blockIdx.x;
    int tid = threadIdx.x;                     // 0..95
    int beg = starts[n];
    int end = starts[n + 1];
    float4 acc = make_float4(0.f, 0.f, 0.f, 0.f);
    const float4* x4 = reinterpret_cast<const float4*>(x);
    for (int j = beg; j < end; ++j) {
        int   s = srcs[j];
        float w = ws[j];
        float4 v = x4[(size_t)s * ROW4 + tid];
        acc.x += w * v.x;
        acc.y += w * v.y;
        acc.z += w * v.z;
        acc.w += w * v.w;
    }
    reinterpret_cast<float4*>(agg)[(size_t)n * ROW4 + tid] = acc;
}

// ---------------------------------------------------------------- layer-1 GEMM
// h1[m][h] = leaky( xT[m] @ W1_self^T + agg1[m] @ W1_neigh^T + b1 ),  m < M1
// M tiles of 16, K=32 (8 k-steps), Nout=64 (4 n-tiles). One wave per M-tile.
__global__ void __launch_bounds__(128)
k_gemm1(const float* __restrict__ xT, const float* __restrict__ agg1,
        const float* __restrict__ Wself, const float* __restrict__ Wneigh,
        const float* __restrict__ bias, float* __restrict__ h1) {
    int lane  = threadIdx.x & 31;
    int wave  = threadIdx.x >> 5;
    int mtile = blockIdx.x * (blockDim.x >> 5) + wave;
    if (mtile >= M1 / 16) return;
    int m0 = mtile * 16;

    v2f aS[8], aN[8];
#pragma unroll
    for (int ks = 0; ks < 8; ++ks) {
        aS[ks] = load_frag(xT,   m0, C_IN, ks * 4, lane);
        aN[ks] = load_frag(agg1, m0, C_IN, ks * 4, lane);
    }

#pragma unroll
    for (int nt = 0; nt < 4; ++nt) {
        int n0 = nt * 16;
        v8f acc = {};
#pragma unroll
        for (int ks = 0; ks < 8; ++ks) {
            v2f bS = load_frag(Wself, n0, C_IN, ks * 4, lane);
            acc = __builtin_amdgcn_wmma_f32_16x16x4_f32(false, aS[ks], false, bS,
                                                        (short)0, acc, false, false);
            v2f bN = load_frag(Wneigh, n0, C_IN, ks * 4, lane);
            acc = __builtin_amdgcn_wmma_f32_16x16x4_f32(false, aN[ks], false, bN,
                                                        (short)0, acc, false, false);
        }
        int col   = n0 + (lane & 15);
        int mbase = m0 + ((lane >> 4) << 3);
        float bv  = bias[col];
#pragma unroll
        for (int r = 0; r < 8; ++r) {
            float v = leaky(acc[r] + bv);
            h1[(size_t)(mbase + r) * H + col] = v;
        }
    }
}

// ---------------------------------------------------------------- z2 GEMM
// z2[m][c] = h1[m] @ W2_neigh^T,  m < M1.  K=64 (16 k-steps), Nout=32 (2 n-tiles).
__global__ void __launch_bounds__(128)
k_gemm_z2(const float* __restrict__ h1, const float* __restrict__ Wneigh2,
          float* __restrict__ z2) {
    int lane  = threadIdx.x & 31;
    int wave  = threadIdx.x >> 5;
    int mtile = blockIdx.x * (blockDim.x >> 5) + wave;
    if (mtile >= M1 / 16) return;
    int m0 = mtile * 16;

    v2f a[16];
#pragma unroll
    for (int ks = 0; ks < 16; ++ks)
        a[ks] = load_frag(h1, m0, H, ks * 4, lane);

#pragma unroll
    for (int nt = 0; nt < 2; ++nt) {
        int n0 = nt * 16;
        v8f acc = {};
#pragma unroll
        for (int ks = 0; ks < 16; ++ks) {
            v2f b = load_frag(Wneigh2, n0, H, ks * 4, lane);
            acc = __builtin_amdgcn_wmma_f32_16x16x4_f32(false, a[ks], false, b,
                                                        (short)0, acc, false, false);
        }
        int col   = n0 + (lane & 15);
        int mbase = m0 + ((lane >> 4) << 3);
#pragma unroll
        for (int r = 0; r < 8; ++r)
            z2[(size_t)(mbase + r) * C_OUT + col] = acc[r];
    }
}

// ---------------------------------------------------------------- final GEMM
// out[t][n][c] = leaky( h1[m] @ W2_self^T + agg2[m][c] + b2[c] ),  m = n*T + t < M2
__global__ void __launch_bounds__(128)
k_final(const float* __restrict__ h1, const float* __restrict__ agg2,
        const float* __restrict__ Wself2, const float* __restrict__ bias2,
        float* __restrict__ out) {
    int lane  = threadIdx.x & 31;
    int wave  = threadIdx.x >> 5;
    int mtile = blockIdx.x * (blockDim.x >> 5) + wave;
    if (mtile >= M2 / 16) return;
    int m0 = mtile * 16;

    v2f a[16];
#pragma unroll
    for (int ks = 0; ks < 16; ++ks)
        a[ks] = load_frag(h1, m0, H, ks * 4, lane);

#pragma unroll
    for (int nt = 0; nt < 2; ++nt) {
        int n0 = nt * 16;
        v8f acc = {};
#pragma unroll
        for (int ks = 0; ks < 16; ++ks) {
            v2f b = load_frag(Wself2, n0, H, ks * 4, lane);
            acc = __builtin_amdgcn_wmma_f32_16x16x4_f32(false, a[ks], false, b,
                                                        (short)0, acc, false, false);
        }
        int col   = n0 + (lane & 15);
        int mbase = m0 + ((lane >> 4) << 3);
        float bv  = bias2[col];
#pragma unroll
        for (int r = 0; r < 8; ++r) {
            int m   = mbase + r;
            float v = leaky(acc[r] + agg2[(size_t)m * C_OUT + col] + bv);
            int n = m / T;
            int t = m % T;
            out[((size_t)t * N_DST2 + n) * C_OUT + col] = v;
        }
    }
}

// ----------------------------------------------------------------
extern "C" void kernel_launch(void* const* d_in, const int* in_sizes, int n_in,
                              void* d_out, int out_size, void* d_ws, size_t ws_size,
                              hipStream_t stream) {
    const float* nf   = (const float*)d_in[0];
    const int*   src1 = (const int*)  d_in[1];
    const int*   dst1 = (const int*)  d_in[2];
    const float* ew1  = (const float*)d_in[3];
    const int*   src2 = (const int*)  d_in[4];
    const int*   dst2 = (const int*)  d_in[5];
    const float* ew2  = (const float*)d_in[6];
    const float* W1s  = (const float*)d_in[7];
    const float* W1n  = (const float*)d_in[8];
    const float* b1   = (const float*)d_in[9];
    const float* W2s  = (const float*)d_in[10];
    const float* W2n  = (const float*)d_in[11];
    const float* b2   = (const float*)d_in[12];
    const int E1 = in_sizes[1];
    const int E2 = in_sizes[4];

    float* out = (float*)d_out;

    // Workspace layout. z2/agg2 reuse xT's region (dead after gemm1/z2 consume xT).
    float* ws   = (float*)d_ws;
    float* xT   = ws;                                    // N_SRC *ROW = 19.20M f
    float* agg1 = xT   + (size_t)N_SRC  * ROW;           // N_DST1*ROW =  7.68M f
    float* h1   = agg1 + (size_t)N_DST1 * ROW;           // M1*H       = 15.36M f
    float* z2   = xT;                                    // N_DST1*ROW =  7.68M f (reuse)
    float* agg2 = xT + (size_t)N_DST1 * ROW;             // N_DST2*ROW =  3.84M f (reuse)
    // CSR scratch after h1:
    int*   ip      = (int*)(h1 + (size_t)M1 * H);
    int*   counts1 = ip;                 ip += N_DST1;
    int*   starts1 = ip;                 ip += N_DST1 + 1;
    int*   cursor1 = ip;                 ip += N_DST1;
    int*   srcs1   = ip;                 ip += E1;
    float* ws1     = (float*)ip;         ip += E1;
    int*   counts2 = ip;                 ip += N_DST2;
    int*   starts2 = ip;                 ip += N_DST2 + 1;
    int*   cursor2 = ip;                 ip += N_DST2;
    int*   srcs2   = ip;                 ip += E2;
    float* ws2     = (float*)ip;

    // 1) transpose node_feat -> xT
    {
        int total = N_SRC * T * (C_IN / 4);
        k_transpose<<<(total + 255) / 256, 256, 0, stream>>>(nf, xT);
    }
    // 2) CSR build for both layers (histogram -> scan -> bin)
    k_zero_int<<<(N_DST1 + 255) / 256, 256, 0, stream>>>(counts1, N_DST1);
    k_zero_int<<<(N_DST2 + 255) / 256, 256, 0, stream>>>(counts2, N_DST2);
    k_hist<<<(E1 + 255) / 256, 256, 0, stream>>>(dst1, counts1, E1);
    k_hist<<<(E2 + 255) / 256, 256, 0, stream>>>(dst2, counts2, E2);
    k_scan<<<1, 1024, 0, stream>>>(counts1, starts1, cursor1, N_DST1);
    k_scan<<<1, 1024, 0, stream>>>(counts2, starts2, cursor2, N_DST2);
    k_bin<<<(E1 + 255) / 256, 256, 0, stream>>>(src1, dst1, ew1, cursor1, srcs1, ws1, E1);
    k_bin<<<(E2 + 255) / 256, 256, 0, stream>>>(src2, dst2, ew2, cursor2, srcs2, ws2, E2);

    // 3) layer-1 aggregation: atomic-free CSR gather (L2-resident xT)
    k_aggregate<<<N_DST1, 96, 0, stream>>>(starts1, srcs1, ws1, xT, agg1);

    // 4) layer-1 fused GEMM (self + neigh) + bias + leaky -> h1
    k_gemm1<<<(M1 / 16 + 3) / 4, 128, 0, stream>>>(xT, agg1, W1s, W1n, b1, h1);

    // 5) pre-apply W2_neigh: z2 = h1 @ W2_neigh^T (aggregate in 32-dim, halves traffic)
    k_gemm_z2<<<(M1 / 16 + 3) / 4, 128, 0, stream>>>(h1, W2n, z2);

    // 6) layer-2 aggregation: CSR gather over z2
    k_aggregate<<<N_DST2, 96, 0, stream>>>(starts2, srcs2, ws2, z2, agg2);

    // 7) final GEMM (self term) + agg2 + bias + leaky + transposed store
    k_final<<<(M2 / 16 + 3) / 4, 128, 0, stream>>>(h1, agg2, W2s, b2, out);
}